// SecondOrder_17557826306743
// MI455X (gfx1250) — compile-verified
//
#include <hip/hip_runtime.h>
#include <stdint.h>

#define FIELDS 50
#define EMBED  64
#define ROWS_PER_BLOCK 16   // 8 waves x 2 batch rows per wave
#define NTHREADS 256

// CDNA5 async global->LDS copy (ASYNCcnt-tracked, gfx1250 path).
__device__ __forceinline__ void async_g2l_b32(void* lds_dst, const void* gsrc) {
    uint32_t l = (uint32_t)(uintptr_t)lds_dst;
    uint64_t g = (uint64_t)(uintptr_t)gsrc;
    asm volatile("global_load_async_to_lds_b32 %0, %1, off"
                 :: "v"(l), "v"(g) : "memory");
}

__device__ __forceinline__ void wait_async0() {
#if __has_builtin(__builtin_amdgcn_s_wait_asynccnt)
    __builtin_amdgcn_s_wait_asynccnt(0);
#else
    asm volatile("s_wait_asynccnt 0" ::: "memory");
#endif
}

__global__ __launch_bounds__(NTHREADS)
void fm_second_order_kernel(const float* __restrict__ vals,
                            const int*   __restrict__ idx,
                            const float* __restrict__ emb,
                            float*       __restrict__ out,
                            int batch)
{
    // Interleaved (idx, value) pairs; hot loops read 2 pairs at a time (ds_load_b128).
    __shared__ __align__(16) int2 s_pair[ROWS_PER_BLOCK * FIELDS];

    const int tid  = threadIdx.x;
    const int wave = tid >> 5;
    const int lane = tid & 31;

    const int block_row0 = blockIdx.x * ROWS_PER_BLOCK;
    int rows = batch - block_row0;
    if (rows > ROWS_PER_BLOCK) rows = ROWS_PER_BLOCK;
    const int    n_stage = rows * FIELDS;           // <= 800 pairs
    const size_t gbase   = (size_t)block_row0 * FIELDS;

    // ---- Stage (idx, value) tile into LDS via CDNA5 async copies (scatter-interleave) ----
    for (int off = tid; off < n_stage; off += NTHREADS) {
        async_g2l_b32(&s_pair[off].x, idx  + gbase + off);
        async_g2l_b32(&s_pair[off].y, vals + gbase + off);
    }
    wait_async0();        // own wave's async LDS writes complete
    __syncthreads();      // cross-wave visibility

    // 16 lanes x float4 cover one 64-wide embedding row; 2 batch rows per wave.
    const int b_local = (wave << 1) | (lane >> 4);
    const int b       = block_row0 + b_local;
    const int kvec    = (lane & 15) << 2;           // float index of this lane's float4
    const uint32_t kbyte = (uint32_t)kvec << 2;     // byte offset within the 256B row
    if (b >= batch) return;

    const int rb = b_local * FIELDS;                // 400B row stride -> 16B aligned
    const char* __restrict__ embb = (const char*)emb;

    // ---- Launch all gathered rows toward L2/L0 (global_prefetch_b8).
    // 32-bit byte offsets (max 1e6*256B < 2^32) -> saddr+voffset addressing.
    #pragma unroll
    for (int f = 0; f < FIELDS; f += 2) {
        const int4 q = *reinterpret_cast<const int4*>(&s_pair[rb + f]);  // ds_load_b128
        __builtin_prefetch(embb + (((uint32_t)q.x << 8) + kbyte), 0, 0);
        __builtin_prefetch(embb + (((uint32_t)q.z << 8) + kbyte), 0, 0);
    }

    // ---- Gather + reduce: a1 = sum(p), a2 = sum(p*p), p = v*e ----
    float a1x = 0.f, a1y = 0.f, a1z = 0.f, a1w = 0.f;
    float a2x = 0.f, a2y = 0.f, a2z = 0.f, a2w = 0.f;

    #pragma unroll 5   // 2 fields per body -> 10 b128 gathers in flight
    for (int f = 0; f < FIELDS; f += 2) {
        const int4 q = *reinterpret_cast<const int4*>(&s_pair[rb + f]);  // ds_load_b128

        const float4 e0 = *reinterpret_cast<const float4*>(
            embb + (((uint32_t)q.x << 8) + kbyte));
        const float4 e1 = *reinterpret_cast<const float4*>(
            embb + (((uint32_t)q.z << 8) + kbyte));

        const float v0 = __int_as_float(q.y);
        const float v1 = __int_as_float(q.w);

        const float p0 = v0 * e0.x, p1 = v0 * e0.y, p2 = v0 * e0.z, p3 = v0 * e0.w;
        a1x += p0; a1y += p1; a1z += p2; a1w += p3;
        a2x = fmaf(p0, p0, a2x);
        a2y = fmaf(p1, p1, a2y);
        a2z = fmaf(p2, p2, a2z);
        a2w = fmaf(p3, p3, a2w);

        const float q0 = v1 * e1.x, q1 = v1 * e1.y, q2 = v1 * e1.z, q3 = v1 * e1.w;
        a1x += q0; a1y += q1; a1z += q2; a1w += q3;
        a2x = fmaf(q0, q0, a2x);
        a2y = fmaf(q1, q1, a2y);
        a2z = fmaf(q2, q2, a2z);
        a2w = fmaf(q3, q3, a2w);
    }

    float4 r;
    r.x = fmaf(a1x, a1x, -a2x);
    r.y = fmaf(a1y, a1y, -a2y);
    r.z = fmaf(a1z, a1z, -a2z);
    r.w = fmaf(a1w, a1w, -a2w);
    *reinterpret_cast<float4*>(out + (size_t)b * EMBED + kvec) = r;
}

extern "C" void kernel_launch(void* const* d_in, const int* in_sizes, int n_in,
                              void* d_out, int out_size, void* d_ws, size_t ws_size,
                              hipStream_t stream) {
    const float* vals = (const float*)d_in[0];   // [B, F] f32
    const int*   idxp = (const int*)d_in[1];     // [B, F] int
    const float* emb  = (const float*)d_in[2];   // [1e6, 64] f32
    float* out = (float*)d_out;                  // [B, 64] f32

    const int batch = in_sizes[0] / FIELDS;
    const int grid  = (batch + ROWS_PER_BLOCK - 1) / ROWS_PER_BLOCK;
    fm_second_order_kernel<<<grid, NTHREADS, 0, stream>>>(vals, idxp, emb, out, batch);
}